// Avg_18700287607025
// MI455X (gfx1250) — compile-verified
//
#include <hip/hip_runtime.h>
#include <hip/hip_bf16.h>

// Ragged segment mean: res[b,d] = sum_{l=begin[b]}^{end[b]-1} seq[b,l,d] / (end[b]-begin[b])
//
// Bandwidth-bound (~0.54 GB useful traffic @ 23.3 TB/s ~= 23 us). Strategy:
//  - one wave32 per (batch, 16-channel tile of D); 32 tiles per batch
//  - iterate only over live rows, 4 rows per V_WMMA_F32_16X16X4_F32
//    (fp32 in / fp32 accumulate => numerically identical to scalar FMA)
//  - peel first/last 4-row chunk (the only ones that can be partial);
//    interior loop uses constant A={1,1} and unconditional clamped loads
//  - divide by count once at the end (matches reference ssum/count numerics)

typedef __attribute__((ext_vector_type(2))) float v2f;
typedef __attribute__((ext_vector_type(8))) float v8f;

#define L_SEQ 512
#define D_DIM 512

__device__ __forceinline__ v8f wmma_acc(v2f a, v2f b, v8f c) {
  return __builtin_amdgcn_wmma_f32_16x16x4_f32(
      /*neg_a=*/false, a, /*neg_b=*/false, b,
      /*c_mod=*/(short)0, c, /*reuse_a=*/false, /*reuse_b=*/false);
}

__global__ __launch_bounds__(256) void ragged_mean_wmma(
    const float* __restrict__ seq,
    const int*   __restrict__ beg,
    const int*   __restrict__ end,
    float*       __restrict__ out) {
  const int lane  = threadIdx.x & 31;
  const int wave  = threadIdx.x >> 5;       // 0..7 within block
  const int b     = blockIdx.x;             // batch
  const int tile  = blockIdx.y * 8 + wave;  // 0..31 : which 16-channel tile
  const int d0    = tile * 16;
  const int n     = lane & 15;              // channel within tile (B/C layout: N = lane)
  const int khalf = (lane >> 4) << 1;       // lanes 0-15 -> K=0,1 ; lanes 16-31 -> K=2,3

  const int bg = beg[b];
  const int en = end[b];

  // column base: &seq[b][0][d0 + n]; rows stride D_DIM floats
  const float* colbase = seq + (size_t)b * L_SEQ * D_DIM + d0 + n;

  v8f c = {};
  const v2f a_one = {1.0f, 1.0f};

  const int l0 = bg & ~3;        // first 4-row chunk (may have leading masked rows)
  const int lN = (en - 1) & ~3;  // last 4-row chunk (may have trailing masked rows)

  {  // first chunk: mask both ends (covers the tiny-segment case l0 == lN)
    const int la = l0 + khalf;
    const int lb = la + 1;
    const float w0 = (la >= bg && la < en) ? 1.0f : 0.0f;
    const float w1 = (lb >= bg && lb < en) ? 1.0f : 0.0f;
    const int ra = (la < en) ? la : en - 1;  // clamp: always in-bounds, weight kills it
    const int rb = (lb < en) ? lb : en - 1;
    v2f bm = {colbase[(size_t)ra * D_DIM], colbase[(size_t)rb * D_DIM]};
    v2f a  = {w0, w1};
    c = wmma_acc(a, bm, c);
  }

  if (lN > l0) {
    // interior chunks (l0+4 .. lN-4): all 4 rows live, A is constant {1,1}
    const float* p = colbase + (size_t)(l0 + 4 + khalf) * D_DIM;
#pragma unroll 4
    for (int l = l0 + 4; l < lN; l += 4) {
      v2f bm = {p[0], p[D_DIM]};  // rows l+khalf, l+khalf+1 (offset:2048 imm)
      c = wmma_acc(a_one, bm, c);
      p += 4 * D_DIM;
    }

    {  // last chunk: rows >= bg guaranteed here, mask only the tail
      const int la = lN + khalf;
      const int lb = la + 1;
      const float w0 = (la < en) ? 1.0f : 0.0f;
      const float w1 = (lb < en) ? 1.0f : 0.0f;
      const int ra = (la < en) ? la : en - 1;
      const int rb = (lb < en) ? lb : en - 1;
      v2f bm = {colbase[(size_t)ra * D_DIM], colbase[(size_t)rb * D_DIM]};
      v2f a  = {w0, w1};
      c = wmma_acc(a, bm, c);
    }
  }

  // C/D layout: VGPR0, lanes 0..15 = row M=0, N=0..15 (all rows identical here)
  if (lane < 16) {
    out[(size_t)b * D_DIM + d0 + n] = c[0] / (float)(en - bg);
  }
}

extern "C" void kernel_launch(void* const* d_in, const int* in_sizes, int n_in,
                              void* d_out, int out_size, void* d_ws, size_t ws_size,
                              hipStream_t stream) {
  const float* seq = (const float*)d_in[0];  // [B, L, D] fp32
  const int*   bg  = (const int*)d_in[1];    // [B] int32
  const int*   en  = (const int*)d_in[2];    // [B] int32
  float*       out = (float*)d_out;          // [B, D] fp32

  const int B = in_sizes[1];                 // 2048
  dim3 grid(B, (D_DIM / 16) / 8);            // (2048, 4): 8 waves/block x 4 = 32 d-tiles
  ragged_mean_wmma<<<grid, 256, 0, stream>>>(seq, bg, en, out);
}